// VectorQuantizer_71889162600504
// MI455X (gfx1250) — compile-verified
//
#include <hip/hip_runtime.h>

typedef float v2f __attribute__((ext_vector_type(2)));
typedef float v8f __attribute__((ext_vector_type(8)));

#define K_CODES 8192
#define DIM     256
#define N_VEC   16384
#define M_TILE  128   // query rows per WG (persistent in LDS)
#define K_CHUNK 64    // codebook rows per chunk
#define DC      32    // d-columns per B stage
#define LDA_A   260   // A row stride (floats): 256+4, stride%64==4 -> conflict-free b64 fetch
#define LDA_B   36    // B row stride (floats): 32+4,  stride%64==36 -> conflict-free b64 fetch

// ---------------- Kernel 1: codebook row norms ||e_k||^2 ----------------
__global__ __launch_bounds__(256) void vq_norms(const float* __restrict__ cb,
                                                float* __restrict__ norms) {
  const int k   = blockIdx.x;
  const int tid = threadIdx.x;
  const float v = cb[(size_t)k * DIM + tid];
  float s = v * v;
  #pragma unroll
  for (int m = 16; m >= 1; m >>= 1) s += __shfl_xor(s, m, 32);
  __shared__ float part[8];
  const int wave = tid >> 5, lane = tid & 31;
  if (lane == 0) part[wave] = s;
  __syncthreads();
  if (tid == 0) {
    float tot = 0.f;
    #pragma unroll
    for (int w = 0; w < 8; ++w) tot += part[w];
    norms[k] = tot;
  }
}

// ---------------- Kernel 2: fused GEMM (f32 WMMA) + per-row argmin ----------------
// 128 WGs x 128 rows. Query tile persistent in LDS; codebook tiles double-buffered
// via async global->LDS (ASYNCcnt) overlapped with WMMA compute.
// Waves: 4 m-groups x 2 k-groups; each wave = 2 m-tiles x 2 k-tiles of 16x16 f32 acc.
__global__ __launch_bounds__(256) void vq_argmin(const float* __restrict__ hs,
                                                 const float* __restrict__ cb,
                                                 const float* __restrict__ norms,
                                                 int* __restrict__ out_idx) {
  __shared__ float Asm[M_TILE * LDA_A];        // 133,120 B: persistent 128 x 256 queries
  __shared__ float Bsm[2 * K_CHUNK * LDA_B];   //  18,432 B: double-buffered 64 x 32 codes
  __shared__ float redV[2 * M_TILE];
  __shared__ int   redI[2 * M_TILE];

  const int tid  = threadIdx.x;
  const int wave = tid >> 5;
  const int lane = tid & 31;
  const int l16  = lane & 15;
  const int lhi  = lane >> 4;
  const int mg   = wave >> 1;   // 0..3 : 32-row group
  const int kg   = wave & 1;    // 0..1 : 32-k group

  const int nbase  = blockIdx.x * M_TILE;
  const int bImg   = nbase >> 10;
  const int hwbase = nbase & 1023;
  const float* hsB = hs + (size_t)bImg * (DIM * 1024) + hwbase;

  // ---- one-time persistent A fill: flat[n][d] = hs[(b*256+d)*1024+hw], n fastest ----
  {
    const int r  = tid & 127;
    const int c0 = tid >> 7;    // 0..1
    #pragma unroll 8
    for (int i = 0; i < 128; ++i) {
      const int c = c0 + (i << 1);
      Asm[r * LDA_A + c] = hsB[((size_t)c << 10) + r];
    }
  }

  const uint32_t bsmBase = (uint32_t)(uintptr_t)(&Bsm[0]);
  // Issue async fills for one 64k x 32d tile: 2 x b128 per thread.
  auto issueB = [&](int buf, int kbase, int dbase) {
    const int r  = tid >> 2;          // 0..63 : codebook row in tile
    const int c0 = (tid & 3) << 2;    // 0,4,8,12 : float column
    const uint64_t g0 =
        (uint64_t)(uintptr_t)(cb + (size_t)(kbase + r) * DIM + dbase);
    const uint32_t l0 =
        bsmBase + (uint32_t)((((buf * K_CHUNK) + r) * LDA_B + c0) * 4);
    asm volatile("global_load_async_to_lds_b128 %0, %1, off"
                 :: "v"(l0), "v"(g0 + (uint64_t)(c0 * 4)) : "memory");
    asm volatile("global_load_async_to_lds_b128 %0, %1, off"
                 :: "v"(l0 + 64u), "v"(g0 + (uint64_t)((c0 + 16) * 4)) : "memory");
  };

  float bestV[2][8];
  int   bestI[2][8];
  #pragma unroll
  for (int t = 0; t < 2; ++t)
    #pragma unroll
    for (int r = 0; r < 8; ++r) { bestV[t][r] = 3.4e38f; bestI[t][r] = 0; }

  __syncthreads();                 // A tile visible to all waves
  issueB(0, 0, 0);                 // preload first B stage

  for (int kc = 0; kc < K_CODES; kc += K_CHUNK) {
    v8f acc[2][2];
    #pragma unroll
    for (int t = 0; t < 2; ++t)
      #pragma unroll
      for (int kt = 0; kt < 2; ++kt)
        #pragma unroll
        for (int i = 0; i < 8; ++i) acc[t][kt][i] = 0.f;

    #pragma unroll
    for (int dc = 0; dc < DIM / DC; ++dc) {     // 8 stages, buffer parity = dc&1
      if (dc < DIM / DC - 1) {
        issueB((dc + 1) & 1, kc, (dc + 1) * DC);
        asm volatile("s_wait_asynccnt 2" ::: "memory");
      } else if (kc + K_CHUNK < K_CODES) {
        issueB(0, kc + K_CHUNK, 0);
        asm volatile("s_wait_asynccnt 2" ::: "memory");
      } else {
        asm volatile("s_wait_asynccnt 0" ::: "memory");
      }
      __syncthreads();             // current B buffer filled by all waves' asyncs

      const float* Arow = &Asm[(mg * 32 + l16) * LDA_A + 2 * lhi + dc * DC];
      const float* Brow = &Bsm[((dc & 1) * K_CHUNK + kg * 32 + l16) * LDA_B + 2 * lhi];
      #pragma unroll
      for (int ds = 0; ds < DC; ds += 4) {
        const v2f b0 = *(const v2f*)(Brow + ds);
        const v2f b1 = *(const v2f*)(Brow + 16 * LDA_B + ds);
        const v2f a0 = *(const v2f*)(Arow + ds);
        const v2f a1 = *(const v2f*)(Arow + 16 * LDA_A + ds);
        acc[0][0] = __builtin_amdgcn_wmma_f32_16x16x4_f32(false, a0, false, b0, (short)0, acc[0][0], false, false);
        acc[0][1] = __builtin_amdgcn_wmma_f32_16x16x4_f32(false, a0, false, b1, (short)0, acc[0][1], false, false);
        acc[1][0] = __builtin_amdgcn_wmma_f32_16x16x4_f32(false, a1, false, b0, (short)0, acc[1][0], false, false);
        acc[1][1] = __builtin_amdgcn_wmma_f32_16x16x4_f32(false, a1, false, b1, (short)0, acc[1][1], false, false);
      }
      __syncthreads();             // all waves done reading this buffer
    }

    // ---- fold chunk into running argmin: dist = ||e||^2 - 2 f.e (k-tile 0 first: ties -> lower k)
    const int   k0 = kc + kg * 32 + l16;
    const float n0 = norms[k0];
    const float n1 = norms[k0 + 16];
    #pragma unroll
    for (int t = 0; t < 2; ++t)
      #pragma unroll
      for (int r = 0; r < 8; ++r) {
        const float d0 = fmaf(-2.0f, acc[t][0][r], n0);
        if (d0 < bestV[t][r]) { bestV[t][r] = d0; bestI[t][r] = k0; }
        const float d1 = fmaf(-2.0f, acc[t][1][r], n1);
        if (d1 < bestV[t][r]) { bestV[t][r] = d1; bestI[t][r] = k0 + 16; }
      }
  }

  // ---- 16-lane min+argmin reduction (k candidates live across lanes of each half)
  #pragma unroll
  for (int t = 0; t < 2; ++t)
    #pragma unroll
    for (int r = 0; r < 8; ++r) {
      float bv = bestV[t][r];
      int   bi = bestI[t][r];
      #pragma unroll
      for (int m = 8; m >= 1; m >>= 1) {
        const float ov = __shfl_xor(bv, m, 32);
        const int   oi = __shfl_xor(bi, m, 32);
        if (ov < bv || (ov == bv && oi < bi)) { bv = ov; bi = oi; }
      }
      if (l16 == 0) {
        const int row = mg * 32 + t * 16 + r + 8 * lhi;   // C/D layout: vgpr r=row r (+8 high half)
        redV[kg * M_TILE + row] = bv;
        redI[kg * M_TILE + row] = bi;
      }
    }
  __syncthreads();
  // ---- merge the two k-groups per row
  if (tid < M_TILE) {
    float bv = redV[tid];
    int   bi = redI[tid];
    const float ov = redV[M_TILE + tid];
    const int   oi = redI[M_TILE + tid];
    if (ov < bv || (ov == bv && oi < bi)) { bv = ov; bi = oi; }
    out_idx[nbase + tid] = bi;
  }
}

// ---------------- Kernel 3: gather + NCHW writeback + per-row sq-err partials ----------------
__global__ __launch_bounds__(256) void vq_gather(const float* __restrict__ hs,
                                                 const float* __restrict__ cb,
                                                 const int* __restrict__ idx,
                                                 float* __restrict__ out_q,
                                                 float* __restrict__ out_idx,
                                                 float* __restrict__ partials) {
  const int n  = blockIdx.x;
  const int c  = threadIdx.x;
  const int b  = n >> 10, hw = n & 1023;
  const int code = idx[n];
  const float q  = cb[(size_t)code * DIM + c];
  const size_t off = ((size_t)(b * DIM + c) << 10) + hw;
  const float x  = hs[off];
  out_q[off] = q;                      // straight-through forward value == quantized
  const float d = q - x;
  float s = d * d;
  #pragma unroll
  for (int m = 16; m >= 1; m >>= 1) s += __shfl_xor(s, m, 32);
  __shared__ float part[8];
  const int wave = c >> 5, lane = c & 31;
  if (lane == 0) part[wave] = s;
  __syncthreads();
  if (c == 0) {
    float tot = 0.f;
    #pragma unroll
    for (int w = 0; w < 8; ++w) tot += part[w];
    partials[n] = tot;
    out_idx[n]  = (float)code;
  }
}

// ---------------- Kernel 4: deterministic loss reduction ----------------
__global__ __launch_bounds__(256) void vq_finalize(const float* __restrict__ partials,
                                                   float* __restrict__ out_loss) {
  const int tid = threadIdx.x;
  float s = 0.f;
  for (int i = tid; i < N_VEC; i += 256) s += partials[i];
  #pragma unroll
  for (int m = 16; m >= 1; m >>= 1) s += __shfl_xor(s, m, 32);
  __shared__ float part[8];
  const int wave = tid >> 5, lane = tid & 31;
  if (lane == 0) part[wave] = s;
  __syncthreads();
  if (tid == 0) {
    float tot = 0.f;
    #pragma unroll
    for (int w = 0; w < 8; ++w) tot += part[w];
    // vq_loss = q_loss + 0.25*e_loss = 1.25 * mean((q-x)^2) over 16*256*32*32 elems
    out_loss[0] = tot * (1.25f / 4194304.0f);
  }
}

extern "C" void kernel_launch(void* const* d_in, const int* in_sizes, int n_in,
                              void* d_out, int out_size, void* d_ws, size_t ws_size,
                              hipStream_t stream) {
  (void)in_sizes; (void)n_in; (void)out_size; (void)ws_size;
  const float* hs = (const float*)d_in[0];   // [16,256,32,32]
  const float* cb = (const float*)d_in[1];   // [8192,256]

  float* out      = (float*)d_out;
  float* out_q    = out;                       // 4,194,304 floats (NCHW quantized)
  float* out_idx  = out + 4194304;             // 16,384 floats (indices)
  float* out_loss = out + 4194304 + 16384;     // 1 float (vq_loss)

  float* ws_norm = (float*)d_ws;                                  // 8192 f32
  int*   ws_idx  = (int*)  ((char*)d_ws + 8192 * 4);              // 16384 i32
  float* ws_part = (float*)((char*)d_ws + 8192 * 4 + 16384 * 4);  // 16384 f32

  vq_norms   <<<K_CODES,        256, 0, stream>>>(cb, ws_norm);
  vq_argmin  <<<N_VEC / M_TILE, 256, 0, stream>>>(hs, cb, ws_norm, ws_idx);
  vq_gather  <<<N_VEC,          256, 0, stream>>>(hs, cb, ws_idx, out_q, out_idx, ws_part);
  vq_finalize<<<1,              256, 0, stream>>>(ws_part, out_loss);
}